// Net_5549097746898
// MI455X (gfx1250) — compile-verified
//
#include <hip/hip_runtime.h>
#include <math.h>

// Problem constants (from reference setup_inputs)
#define Bsz     16
#define Npts    4096
#define Hdim    128
#define Kn      10
#define NLAYERS 4
#define BN_EPS  1e-5f

typedef float v2f __attribute__((ext_vector_type(2)));
typedef float v8f __attribute__((ext_vector_type(8)));

// ------------------------------------------------------------------
// 1. kNN: per batch, stage all 4096 points (48KB) in LDS; each thread
//    owns one query and keeps a sorted top-10 in registers.
// ------------------------------------------------------------------
__global__ __launch_bounds__(256) void knn_kernel(const float* __restrict__ pos,
                                                  int* __restrict__ nbr)
{
    __shared__ float sx[Npts], sy[Npts], sz[Npts];
    const int b     = blockIdx.x >> 4;     // 16 chunks per batch
    const int chunk = blockIdx.x & 15;
    const float* pb = pos + (size_t)b * Npts * 3;
    for (int i = threadIdx.x; i < Npts * 3; i += 256) {
        float v = pb[i];
        int n = i / 3, d = i - n * 3;
        if (d == 0)      sx[n] = v;
        else if (d == 1) sy[n] = v;
        else             sz[n] = v;
    }
    __syncthreads();

    const int q = chunk * 256 + threadIdx.x;
    const float qx = sx[q], qy = sy[q], qz = sz[q];

    float bd[Kn]; int bi[Kn];
#pragma unroll
    for (int s = 0; s < Kn; ++s) { bd[s] = 3.4e38f; bi[s] = 0; }

    for (int j = 0; j < Npts; ++j) {
        float dx = qx - sx[j], dy = qy - sy[j], dz = qz - sz[j];
        float d2 = dx * dx + dy * dy + dz * dz;
        if (d2 < bd[Kn - 1]) {
            float dv = d2; int iv = j;
#pragma unroll
            for (int s = 0; s < Kn; ++s) {
                if (dv < bd[s]) {
                    float td = bd[s]; int ti = bi[s];
                    bd[s] = dv; bi[s] = iv;
                    dv = td; iv = ti;
                }
            }
        }
    }
    int* o = nbr + ((size_t)b * Npts + q) * Kn;
#pragma unroll
    for (int s = 0; s < Kn; ++s) o[s] = bi[s];
}

// ------------------------------------------------------------------
// 2. x0 = relu(pos @ W0 + b0)   (K=3 — plain VALU, thread = node x 4ch)
// ------------------------------------------------------------------
__global__ __launch_bounds__(256) void x0_kernel(const float* __restrict__ pos,
                                                 const float* __restrict__ W0,
                                                 const float* __restrict__ b0,
                                                 float* __restrict__ x0)
{
    int t    = blockIdx.x * 256 + threadIdx.x;   // [0, B*N*32)
    int h4   = (t & 31) << 2;
    int node = t >> 5;
    const float* p = pos + (size_t)node * 3;
    float px = p[0], py = p[1], pz = p[2];
    float* o = x0 + (size_t)node * Hdim + h4;
#pragma unroll
    for (int u = 0; u < 4; ++u) {
        int h = h4 + u;
        float r = px * W0[h] + py * W0[Hdim + h] + pz * W0[2 * Hdim + h] + b0[h];
        o[u] = fmaxf(r, 0.0f);
    }
}

// ------------------------------------------------------------------
// 3. h = 0.9 * sum_k x[nbr] + 0.1 * x0   (float4 gathers)
// ------------------------------------------------------------------
__global__ __launch_bounds__(256) void gather_kernel(const float* __restrict__ x,
                                                     const float* __restrict__ x0,
                                                     const int* __restrict__ nbr,
                                                     float* __restrict__ hbuf)
{
    int t    = blockIdx.x * 256 + threadIdx.x;   // [0, B*N*32)
    int h4   = (t & 31) << 2;
    int node = t >> 5;
    int b    = node >> 12;                       // N = 4096
    const int* nb = nbr + (size_t)node * Kn;
    float s0 = 0.f, s1 = 0.f, s2 = 0.f, s3 = 0.f;
#pragma unroll
    for (int k = 0; k < Kn; ++k) {
        int j = nb[k];
        const float4 xv = *reinterpret_cast<const float4*>(
            x + ((size_t)(b << 12) + j) * Hdim + h4);
        s0 += xv.x; s1 += xv.y; s2 += xv.z; s3 += xv.w;
    }
    const float4 x0v = *reinterpret_cast<const float4*>(x0 + (size_t)node * Hdim + h4);
    float4 r;
    r.x = 0.9f * s0 + 0.1f * x0v.x;
    r.y = 0.9f * s1 + 0.1f * x0v.y;
    r.z = 0.9f * s2 + 0.1f * x0v.z;
    r.w = 0.9f * s3 + 0.1f * x0v.w;
    *reinterpret_cast<float4*>(hbuf + (size_t)node * Hdim + h4) = r;
}

// ------------------------------------------------------------------
// 4. WMMA GEMM (f32, V_WMMA_F32_16X16X4_F32):
//    out = [resid*H + scale*(H@W) (+bias)] (relu optional)
//    M=65536, N=K=128. Block = 256 thr (8 waves); wave owns 16 rows x
//    128 cols = 8 accumulators.
//    W staged in LDS in K-PAIR-MAJOR layout:
//       sW[(k/2)*PSTRIDE + col*2 + (k&1)]
//    -> each B fragment is one contiguous 8B element (ds_load_b64 /
//    ds_load_2addr_b64, conflict-free with PSTRIDE=288).
//    All 8 B fragments of a k-step are loaded BEFORE the 8 WMMAs so
//    the DS pipe runs ahead of the matrix pipe (one DScnt drain per
//    k-step instead of four).
// ------------------------------------------------------------------
#define PSTRIDE 288

__global__ __launch_bounds__(256) void gcn_gemm_kernel(
    const float* __restrict__ Hm, const float* __restrict__ Wl,
    const float* __restrict__ bias, float* __restrict__ Xout,
    float resid, float scale, int hasBias, int doRelu)
{
    __shared__ float sW[32 * PSTRIDE];             // 36 KB
    const int lane    = threadIdx.x & 31;
    const int wave    = threadIdx.x >> 5;
    const int l15     = lane & 15;
    const int hsel    = lane >> 4;                 // lane-half select
    const int rowBase = blockIdx.x * 128 + wave * 16;

    v8f acc[8];
#pragma unroll
    for (int t = 0; t < 8; ++t)
        acc[t] = (v8f){0.f, 0.f, 0.f, 0.f, 0.f, 0.f, 0.f, 0.f};

    // A fragment source: row = rowBase + l15, K offset hsel*2 (ISA 16x4 layout)
    const float* aRow = Hm + (size_t)(rowBase + l15) * Hdim + (hsel << 1);

    for (int kh = 0; kh < 2; ++kh) {
        __syncthreads();
        for (int i = threadIdx.x; i < 64 * 128; i += 256) {
            int r = i >> 7, c = i & 127;           // k-row in half, col
            sW[(r >> 1) * PSTRIDE + (c << 1) + (r & 1)] = Wl[(kh << 13) + i];
        }
        __syncthreads();
#pragma unroll 4
        for (int ks = 0; ks < 64; ks += 4) {
            const float2 av = *reinterpret_cast<const float2*>(aRow + (kh << 6) + ks);
            v2f a; a.x = av.x; a.y = av.y;
            // Batch-load all 8 B fragments for this k-step, then WMMA.
            const float* bBase = &sW[((ks >> 1) + hsel) * PSTRIDE + (l15 << 1)];
            v2f bf[8];
#pragma unroll
            for (int t = 0; t < 8; ++t) {
                const float2 bv = *reinterpret_cast<const float2*>(bBase + t * 32);
                bf[t].x = bv.x; bf[t].y = bv.y;
            }
#pragma unroll
            for (int t = 0; t < 8; ++t) {
                acc[t] = __builtin_amdgcn_wmma_f32_16x16x4_f32(
                    false, a, false, bf[t], (short)0, acc[t], false, false);
            }
        }
    }

    // Epilogue: C/D layout — VGPR v: M = rowBase + v + hsel*8, N = t*16 + l15
#pragma unroll
    for (int t = 0; t < 8; ++t) {
        int n = t * 16 + l15;
#pragma unroll
        for (int v = 0; v < 8; ++v) {
            int m = rowBase + v + (hsel << 3);
            float hv = Hm[(size_t)m * Hdim + n];
            float r  = resid * hv + scale * acc[t][v];
            if (hasBias) r += bias[n];
            if (doRelu)  r = fmaxf(r, 0.0f);
            Xout[(size_t)m * Hdim + n] = r;
        }
    }
}

// ------------------------------------------------------------------
// 5. global max pool over N  (thread per (b,h))
// ------------------------------------------------------------------
__global__ __launch_bounds__(256) void pool_kernel(const float* __restrict__ xin,
                                                   float* __restrict__ pooled)
{
    int t  = blockIdx.x * 256 + threadIdx.x;    // 2048 threads
    int b  = t >> 7, hh = t & 127;
    const float* p = xin + (size_t)b * Npts * Hdim + hh;
    float m = -3.4e38f;
    for (int n = 0; n < Npts; ++n) m = fmaxf(m, p[(size_t)n * Hdim]);
    pooled[t] = m;
}

// ------------------------------------------------------------------
// 6. head: BN(train-stat)+ReLU MLP x2 -> Wout -> log_softmax.
//    Tiny (16 rows) — single block with LDS barriers.
// ------------------------------------------------------------------
__global__ __launch_bounds__(256) void head_kernel(
    const float* __restrict__ pooled,
    const float* __restrict__ Wm1, const float* __restrict__ bm1,
    const float* __restrict__ g1,  const float* __restrict__ be1,
    const float* __restrict__ Wm2, const float* __restrict__ bm2,
    const float* __restrict__ g2,  const float* __restrict__ be2,
    const float* __restrict__ Wout, const float* __restrict__ bout,
    float* __restrict__ out)
{
    __shared__ float yA[Bsz * Hdim];
    __shared__ float yB[Bsz * Hdim];
    __shared__ float z[Bsz * 10];
    const int tid = threadIdx.x;

    // linear 1
    for (int o = tid; o < Bsz * Hdim; o += 256) {
        int b = o >> 7, hh = o & 127;
        float s = bm1[hh];
        for (int k = 0; k < Hdim; ++k) s += pooled[b * Hdim + k] * Wm1[k * Hdim + hh];
        yA[o] = s;
    }
    __syncthreads();
    // BN1 + relu (biased batch stats over B=16)
    if (tid < Hdim) {
        float mu = 0.f;
        for (int b = 0; b < Bsz; ++b) mu += yA[b * Hdim + tid];
        mu *= (1.0f / Bsz);
        float var = 0.f;
        for (int b = 0; b < Bsz; ++b) { float d = yA[b * Hdim + tid] - mu; var += d * d; }
        var *= (1.0f / Bsz);
        float sc = g1[tid] * rsqrtf(var + BN_EPS);
        float sh = be1[tid] - mu * sc;
        for (int b = 0; b < Bsz; ++b)
            yA[b * Hdim + tid] = fmaxf(yA[b * Hdim + tid] * sc + sh, 0.f);
    }
    __syncthreads();
    // linear 2
    for (int o = tid; o < Bsz * Hdim; o += 256) {
        int b = o >> 7, hh = o & 127;
        float s = bm2[hh];
        for (int k = 0; k < Hdim; ++k) s += yA[b * Hdim + k] * Wm2[k * Hdim + hh];
        yB[o] = s;
    }
    __syncthreads();
    // BN2 + relu
    if (tid < Hdim) {
        float mu = 0.f;
        for (int b = 0; b < Bsz; ++b) mu += yB[b * Hdim + tid];
        mu *= (1.0f / Bsz);
        float var = 0.f;
        for (int b = 0; b < Bsz; ++b) { float d = yB[b * Hdim + tid] - mu; var += d * d; }
        var *= (1.0f / Bsz);
        float sc = g2[tid] * rsqrtf(var + BN_EPS);
        float sh = be2[tid] - mu * sc;
        for (int b = 0; b < Bsz; ++b)
            yB[b * Hdim + tid] = fmaxf(yB[b * Hdim + tid] * sc + sh, 0.f);
    }
    __syncthreads();
    // output linear
    if (tid < Bsz * 10) {
        int b = tid / 10, c = tid - b * 10;
        float s = bout[c];
        for (int k = 0; k < Hdim; ++k) s += yB[b * Hdim + k] * Wout[k * 10 + c];
        z[tid] = s;
    }
    __syncthreads();
    // log_softmax
    if (tid < Bsz) {
        float mx = -3.4e38f;
        for (int c = 0; c < 10; ++c) mx = fmaxf(mx, z[tid * 10 + c]);
        float se = 0.f;
        for (int c = 0; c < 10; ++c) se += expf(z[tid * 10 + c] - mx);
        float lse = mx + logf(se);
        for (int c = 0; c < 10; ++c) out[tid * 10 + c] = z[tid * 10 + c] - lse;
    }
}

// ------------------------------------------------------------------
// launch
// ------------------------------------------------------------------
extern "C" void kernel_launch(void* const* d_in, const int* in_sizes, int n_in,
                              void* d_out, int out_size, void* d_ws, size_t ws_size,
                              hipStream_t stream)
{
    const float* pos  = (const float*)d_in[0];
    const float* W0   = (const float*)d_in[1];
    const float* b0   = (const float*)d_in[2];
    const float* Wc   = (const float*)d_in[3];
    const float* W1   = (const float*)d_in[4];
    const float* b1   = (const float*)d_in[5];
    const float* Wm1  = (const float*)d_in[6];
    const float* bm1  = (const float*)d_in[7];
    const float* g1   = (const float*)d_in[8];
    const float* be1  = (const float*)d_in[9];
    const float* Wm2  = (const float*)d_in[10];
    const float* bm2  = (const float*)d_in[11];
    const float* g2   = (const float*)d_in[12];
    const float* be2  = (const float*)d_in[13];
    const float* Wout = (const float*)d_in[14];
    const float* bout = (const float*)d_in[15];
    float* out        = (float*)d_out;

    // workspace layout (~104 MB)
    const size_t XBYTES = (size_t)Bsz * Npts * Hdim * sizeof(float);   // 33.5 MB
    char*  ws   = (char*)d_ws;
    size_t off  = 0;
    int*   nbr  = (int*)(ws + off); off += ((size_t)Bsz * Npts * Kn * sizeof(int) + 255) & ~(size_t)255;
    float* x0b  = (float*)(ws + off); off += XBYTES;
    float* xA   = (float*)(ws + off); off += XBYTES;
    float* xB   = (float*)(ws + off); off += XBYTES;
    float* pooled = (float*)(ws + off); off += (size_t)Bsz * Hdim * sizeof(float);
    (void)ws_size; (void)in_sizes; (void)n_in; (void)out_size;

    const int nodeThreads = Bsz * Npts * (Hdim / 4);      // 2,097,152
    const int nodeBlocks  = nodeThreads / 256;            // 8192
    const int gemmBlocks  = (Bsz * Npts) / 128;           // 512

    // 1. kNN
    knn_kernel<<<Bsz * (Npts / 256), 256, 0, stream>>>(pos, nbr);
    // 2. x0
    x0_kernel<<<nodeBlocks, 256, 0, stream>>>(pos, W0, b0, x0b);

    // 3/4. GCN2Conv layers (ping-pong xA/xB; h always in xB)
    const float* xcur = x0b;
    for (int l = 0; l < NLAYERS; ++l) {
        gather_kernel<<<nodeBlocks, 256, 0, stream>>>(xcur, x0b, nbr, xB);
        float beta = logf(0.5f / (float)(l + 1) + 1.0f);
        gcn_gemm_kernel<<<gemmBlocks, 256, 0, stream>>>(
            xB, Wc + (size_t)l * Hdim * Hdim, (const float*)nullptr, xA,
            1.0f - beta, beta, 0, 1);
        xcur = xA;
    }

    // lins[1]: x @ W1 + b1  (resid=0, scale=1, bias, no relu)
    gcn_gemm_kernel<<<gemmBlocks, 256, 0, stream>>>(
        xA, W1, b1, xB, 0.0f, 1.0f, 1, 0);

    // 5. global max pool
    pool_kernel<<<(Bsz * Hdim) / 256, 256, 0, stream>>>(xB, pooled);

    // 6. head
    head_kernel<<<1, 256, 0, stream>>>(pooled, Wm1, bm1, g1, be1,
                                       Wm2, bm2, g2, be2, Wout, bout, out);
}